// SparseRoutedMLP_21122649162411
// MI455X (gfx1250) — compile-verified
//
#include <hip/hip_runtime.h>
#include <math.h>

typedef float v2f __attribute__((ext_vector_type(2)));
typedef float v8f __attribute__((ext_vector_type(8)));

#define D_MODEL 4096
#define H_DIM   14336
#define N_FEAT  56
#define N_SEL   3584            // 64 * 56 selected rows
#define T_TOK   32
#define KSTEPS1 (D_MODEL / 4)   // 1024 k-steps of 4 for gate/up GEMM
#define KSTEPS2 (N_SEL / 4)     // 896 k-steps of 4 for down GEMM
#define NWAVES  8               // waves per block (256 threads)

__device__ __forceinline__ v8f wmma_f32x4(v2f a, v2f b, v8f c) {
  // D = A(16x4 f32) * B(4x16 f32) + C(16x16 f32)
  return __builtin_amdgcn_wmma_f32_16x16x4_f32(
      /*neg_a=*/false, a, /*neg_b=*/false, b,
      /*c_mod=*/(short)0, c, /*reuse_a=*/false, /*reuse_b=*/false);
}

// ---------------------------------------------------------------------------
// Phase 1: z[t, r] = silu(x @ gate_sel^T) * (x @ up_sel^T) * mask_sel
// One block (8 waves) per 16-row n-tile; each wave covers a K-chunk of 512.
// Weight rows are streamed exactly once from HBM.
// ---------------------------------------------------------------------------
__global__ __launch_bounds__(256) void sparse_mlp_phase1(
    const float* __restrict__ x,        // [32, 4096]
    const float* __restrict__ gate_w,   // [14336, 4096]
    const float* __restrict__ up_w,     // [14336, 4096]
    const float* __restrict__ mask,     // [256]
    const int*   __restrict__ positions,// [64]
    float* __restrict__ z)              // [32, 3584] scratch
{
  __shared__ float red[NWAVES - 1][4][8][32];  // waves 1..7, {g0,g1,u0,u1}, vgpr, lane

  const int tid  = threadIdx.x;
  const int lane = tid & 31;
  const int wave = tid >> 5;
  const int half = lane >> 4;           // 0: K={0,1}, 1: K={2,3}
  const int l16  = lane & 15;

  const int tile = blockIdx.x;          // 0..223
  const int r    = tile * 16 + l16;     // selected column index 0..3583
  const int s    = r / N_FEAT;
  const int f    = r - s * N_FEAT;
  const int pos  = positions[s];
  const float mval = mask[pos];
  const long  grow = (long)pos * N_FEAT + f;   // gathered weight row

  const int kchunk = KSTEPS1 / NWAVES;         // 128 k-steps per wave
  const int kstart = wave * kchunk;

  const float* gp  = gate_w + grow * D_MODEL + (size_t)kstart * 4 + half * 2;
  const float* upp = up_w   + grow * D_MODEL + (size_t)kstart * 4 + half * 2;
  const float* xp0 = x + (size_t)l16 * D_MODEL        + (size_t)kstart * 4 + half * 2;
  const float* xp1 = x + (size_t)(16 + l16) * D_MODEL + (size_t)kstart * 4 + half * 2;

  v8f cg0 = {}, cg1 = {}, cu0 = {}, cu1 = {};

#pragma unroll 4
  for (int kk = 0; kk < kchunk; ++kk) {
    if ((kk & 3) == 0) {                       // once per 64B line per stream
      __builtin_prefetch(gp + 64, 0, 0);       // 256 B ahead in this lane's row
      __builtin_prefetch(upp + 64, 0, 0);
    }
    v2f a0 = *(const v2f*)xp0;
    v2f a1 = *(const v2f*)xp1;
    v2f bg = *(const v2f*)gp;
    v2f bu = *(const v2f*)upp;
    cg0 = wmma_f32x4(a0, bg, cg0);
    cg1 = wmma_f32x4(a1, bg, cg1);
    cu0 = wmma_f32x4(a0, bu, cu0);
    cu1 = wmma_f32x4(a1, bu, cu1);
    xp0 += 4; xp1 += 4; gp += 4; upp += 4;
  }

  if (wave > 0) {
#pragma unroll
    for (int v = 0; v < 8; ++v) {
      red[wave - 1][0][v][lane] = cg0[v];
      red[wave - 1][1][v][lane] = cg1[v];
      red[wave - 1][2][v][lane] = cu0[v];
      red[wave - 1][3][v][lane] = cu1[v];
    }
  }
  __syncthreads();

  if (wave == 0) {
#pragma unroll
    for (int w = 0; w < NWAVES - 1; ++w) {
#pragma unroll
      for (int v = 0; v < 8; ++v) {
        cg0[v] += red[w][0][v][lane];
        cg1[v] += red[w][1][v][lane];
        cu0[v] += red[w][2][v][lane];
        cu1[v] += red[w][3][v][lane];
      }
    }
    // silu(g) * u * mask ; C/D layout: lane holds N=l16, rows M = v + half*8
#pragma unroll
    for (int v = 0; v < 8; ++v) {
      int m0 = v + half * 8;         // token tile 0
      int m1 = 16 + v + half * 8;    // token tile 1
      float g0 = cg0[v], g1 = cg1[v];
      float s0 = g0 / (1.0f + __expf(-g0));
      float s1 = g1 / (1.0f + __expf(-g1));
      z[(size_t)m0 * N_SEL + r] = s0 * cu0[v] * mval;
      z[(size_t)m1 * N_SEL + r] = s1 * cu1[v] * mval;
    }
  }
}

// ---------------------------------------------------------------------------
// Phase 2: out[t, d] = sum_r z[t, r] * down_w[d, row_idx[r]]
// One block (8 waves) per 16-col d-tile; each wave covers 112 k-steps.
// Gather bases (groups of 4 consecutive r share one slice since 4 | 56)
// are precomputed into LDS.
// ---------------------------------------------------------------------------
__global__ __launch_bounds__(256) void sparse_mlp_phase2(
    const float* __restrict__ z,        // [32, 3584]
    const float* __restrict__ down_w,   // [4096, 14336]
    const int*   __restrict__ positions,// [64]
    float* __restrict__ out)            // [32, 4096]
{
  __shared__ int   hgroup[KSTEPS2];     // 896 gather bases
  __shared__ float red[NWAVES - 1][2][8][32];

  const int tid = threadIdx.x;
#pragma unroll 1
  for (int j = tid; j < KSTEPS2; j += 256) {   // wave-uniform trip counts
    int rr = j * 4;
    int s  = rr / N_FEAT;
    int f  = rr - s * N_FEAT;
    hgroup[j] = positions[s] * N_FEAT + f;
  }
  __syncthreads();

  const int lane = tid & 31;
  const int wave = tid >> 5;
  const int half = lane >> 4;
  const int l16  = lane & 15;

  const int tile = blockIdx.x;          // 0..255
  const int d    = tile * 16 + l16;

  const float* dwbase = down_w + (size_t)d * H_DIM + half * 2;
  const int kchunk = KSTEPS2 / NWAVES;  // 112 k-steps per wave
  const int kstart = wave * kchunk;

  const float* zp0 = z + (size_t)l16 * N_SEL        + (size_t)kstart * 4 + half * 2;
  const float* zp1 = z + (size_t)(16 + l16) * N_SEL + (size_t)kstart * 4 + half * 2;

  v8f c0 = {}, c1 = {};

#pragma unroll 4
  for (int kk = kstart; kk < kstart + kchunk; ++kk) {
    int h  = hgroup[kk];
    if ((kk & 3) == 0) {
      __builtin_prefetch(dwbase + h + 64, 0, 0);
    }
    v2f a0 = *(const v2f*)zp0;
    v2f a1 = *(const v2f*)zp1;
    v2f b  = *(const v2f*)(dwbase + h);
    c0 = wmma_f32x4(a0, b, c0);
    c1 = wmma_f32x4(a1, b, c1);
    zp0 += 4; zp1 += 4;
  }

  if (wave > 0) {
#pragma unroll
    for (int v = 0; v < 8; ++v) {
      red[wave - 1][0][v][lane] = c0[v];
      red[wave - 1][1][v][lane] = c1[v];
    }
  }
  __syncthreads();

  if (wave == 0) {
#pragma unroll
    for (int w = 0; w < NWAVES - 1; ++w) {
#pragma unroll
      for (int v = 0; v < 8; ++v) {
        c0[v] += red[w][0][v][lane];
        c1[v] += red[w][1][v][lane];
      }
    }
#pragma unroll
    for (int v = 0; v < 8; ++v) {
      int m0 = v + half * 8;
      int m1 = 16 + v + half * 8;
      out[(size_t)m0 * D_MODEL + d] = c0[v];
      out[(size_t)m1 * D_MODEL + d] = c1[v];
    }
  }
}

extern "C" void kernel_launch(void* const* d_in, const int* in_sizes, int n_in,
                              void* d_out, int out_size, void* d_ws, size_t ws_size,
                              hipStream_t stream) {
  const float* x         = (const float*)d_in[0];
  const float* gate_w    = (const float*)d_in[1];
  const float* up_w      = (const float*)d_in[2];
  const float* down_w    = (const float*)d_in[3];
  const float* mask      = (const float*)d_in[4];
  const int*   positions = (const int*)d_in[5];
  float* out = (float*)d_out;
  float* z   = (float*)d_ws;            // 32*3584*4 = 448 KB scratch

  sparse_mlp_phase1<<<dim3(N_SEL / 16), dim3(256), 0, stream>>>(
      x, gate_w, up_w, mask, positions, z);
  sparse_mlp_phase2<<<dim3(D_MODEL / 16), dim3(256), 0, stream>>>(
      z, down_w, positions, out);
}